// DiffGaussianRasterizer_8134668059150
// MI455X (gfx1250) — compile-verified
//
#include <hip/hip_runtime.h>
#include <hip/hip_bf16.h>

typedef __attribute__((ext_vector_type(2))) float v2f;
typedef __attribute__((ext_vector_type(8))) float v8f;

#define IMG_H 256
#define IMG_W 256
#define NGAUSS 512
#define FOCAL_F 50.0f

// Workspace layout (in floats):
//   gp4  : [0,     2048)  float4 per sorted gaussian: {proj_u, proj_v, 1/su^2, 1/sv^2}
//   gop  : [2048,  2560)  sorted opacity
//   bmat : [2560, 10752)  per (chunk=128, lane=32) v2f B-matrix fragments in WMMA layout
#define WS_GOP 2048
#define WS_BMAT 2560

// Cross-half lane swap (lane i <-> lane i^16) on wave32 without touching LDS:
// v_permlanex16_b32 with identity selects. Pure VALU, no dscnt waits.
__device__ __forceinline__ float lane_xor16(float x) {
    int v = __builtin_bit_cast(int, x);
    int r = __builtin_amdgcn_permlanex16(v, v, 0x76543210, 0xfedcba98, false, false);
    return __builtin_bit_cast(float, r);
}

// ---------------------------------------------------------------------------
// Kernel 1: project gaussians, stable rank-sort by clipped z (O(N^2) via LDS),
// scatter sorted params to ws, and pre-bake the WMMA B-matrix fragments
// (colors of 4 gaussians laid out for V_WMMA_F32_16X16X4_F32, cols 3..15 = 0).
// Single workgroup of 512 threads.
// ---------------------------------------------------------------------------
__global__ __launch_bounds__(512)
void dgr_prep_kernel(const float* __restrict__ means3d,
                     const float* __restrict__ scales,
                     const float* __restrict__ opac,
                     const float* __restrict__ colors,
                     float* __restrict__ ws) {
    __shared__ float zsh[NGAUSS];
    __shared__ float csh[NGAUSS * 3];

    const int i = threadIdx.x;

    const float z = fmaxf(means3d[i * 3 + 2], 0.1f);
    zsh[i] = z;
    __syncthreads();

    // stable argsort rank: count strictly-smaller, tie-break by original index
    int rank = 0;
    for (int j = 0; j < NGAUSS; ++j) {
        const float zj = zsh[j];
        rank += ((zj < z) || (zj == z && j < i)) ? 1 : 0;
    }

    const float pu = FOCAL_F * means3d[i * 3 + 0] / z;
    const float pv = FOCAL_F * means3d[i * 3 + 1] / z;
    const float su = fmaxf(FOCAL_F * scales[i * 3 + 0] / z, 0.5f);
    const float sv = fmaxf(FOCAL_F * scales[i * 3 + 1] / z, 0.5f);

    float4 g;
    g.x = pu;
    g.y = pv;
    g.z = 1.0f / (su * su);
    g.w = 1.0f / (sv * sv);
    ((float4*)ws)[rank] = g;
    ws[WS_GOP + rank] = opac[i];

    csh[rank * 3 + 0] = colors[i * 3 + 0];
    csh[rank * 3 + 1] = colors[i * 3 + 1];
    csh[rank * 3 + 2] = colors[i * 3 + 2];
    __syncthreads();

    // B fragment layout for f32 16x16x4 WMMA (wave32):
    //   VGPR0: lanes 0-15 = row K=0, lanes 16-31 = row K=2
    //   VGPR1: lanes 0-15 = row K=1, lanes 16-31 = row K=3
    // lane%16 = column N (channel; 0..2 = RGB, 3..15 = zero padding)
    float* bmat = ws + WS_BMAT;
    for (int e = i; e < 128 * 32; e += 512) {
        const int t = e >> 5;        // K-chunk index (4 gaussians per chunk)
        const int l = e & 31;        // lane
        const int col = l & 15;
        const int k0 = (l < 16) ? 0 : 2;
        const int n0 = 4 * t;
        float b0 = 0.0f, b1 = 0.0f;
        if (col < 3) {
            b0 = csh[(n0 + k0) * 3 + col];
            b1 = csh[(n0 + k0 + 1) * 3 + col];
        }
        bmat[e * 2 + 0] = b0;
        bmat[e * 2 + 1] = b1;
    }
}

// ---------------------------------------------------------------------------
// Kernel 2: rasterize. One wave32 per 32 pixels. Each lane keeps its pixel's
// transmittance T in a register and evaluates 4 gaussians per chunk
// (sequential compositing order preserved). The alpha*T weights are swapped
// across wave halves with v_permlanex16 into the 16x4 f32 A-matrix layout and
// two v_wmma_f32_16x16x4_f32 ops per chunk accumulate the
// (pixels x gaussians) x (gaussians x rgb) product.
// ---------------------------------------------------------------------------
__global__ __launch_bounds__(256)
void dgr_raster_kernel(const float* __restrict__ ws, float* __restrict__ out) {
    const float4* __restrict__ gp4 = (const float4*)ws;
    const float* __restrict__ gop = ws + WS_GOP;
    const v2f* __restrict__ bmat = (const v2f*)(ws + WS_BMAT);

    const int tid = threadIdx.x;
    const int lane = tid & 31;
    const int wave = tid >> 5;
    const int pixBase = blockIdx.x * 256 + wave * 32;
    const int p = pixBase + lane;

    // meshgrid 'xy': u varies along W (fast axis), v along H
    const float u = (float)(p & (IMG_W - 1)) - (float)(IMG_W / 2);
    const float v = (float)(p >> 8) - (float)(IMG_H / 2);

    const bool lo = (lane < 16);

    float T = 1.0f;
    v8f acc1 = {0.f, 0.f, 0.f, 0.f, 0.f, 0.f, 0.f, 0.f}; // pixels pixBase+0..15
    v8f acc2 = {0.f, 0.f, 0.f, 0.f, 0.f, 0.f, 0.f, 0.f}; // pixels pixBase+16..31

    for (int t = 0; t < NGAUSS / 4; ++t) {
        // keep the per-lane B fragment stream resident ahead of use
        const int pt = (t + 16 < 128) ? (t + 16) : t;
        __builtin_prefetch((const void*)(bmat + pt * 32 + lane), 0, 3);

        float wk[4];
#pragma unroll
        for (int k = 0; k < 4; ++k) {
            const int n = 4 * t + k;
            const float4 g = gp4[n];   // uniform across wave -> scalar loads
            const float op = gop[n];
            const float du = u - g.x;
            const float dv = v - g.y;
            const float e = __expf(-0.5f * (du * du * g.z + dv * dv * g.w));
            const float a = fminf(e * op, 1.0f);   // alpha in [0,1]
            wk[k] = a * T;                          // contribution weight
            T *= (1.0f - a);                        // transmittance update
        }

        // Build A fragments (16 pixels x 4 gaussians, f32):
        //   acc1 tile: M = pixel 0..15  -> lanes 0-15 own K0/K1, lanes 16-31 need K2/K3 of pixel lane-16
        //   acc2 tile: M = pixel 16..31 -> lanes 0-15 need K0/K1 of pixel lane+16, lanes 16-31 own K2/K3
        const float s0 = lane_xor16(wk[0]);
        const float s1 = lane_xor16(wk[1]);
        const float s2 = lane_xor16(wk[2]);
        const float s3 = lane_xor16(wk[3]);

        v2f a1, a2;
        a1.x = lo ? wk[0] : s2;
        a1.y = lo ? wk[1] : s3;
        a2.x = lo ? s0 : wk[2];
        a2.y = lo ? s1 : wk[3];

        const v2f b = bmat[t * 32 + lane];

        acc1 = __builtin_amdgcn_wmma_f32_16x16x4_f32(
            false, a1, false, b, (short)0, acc1, false, false);
        acc2 = __builtin_amdgcn_wmma_f32_16x16x4_f32(
            false, a2, false, b, (short)0, acc2, false, false);
    }

    // D layout: VGPR r, lanes 0-15 -> M=r, lanes 16-31 -> M=r+8; lane%16 = N (channel)
    const int ch = lane & 15;
    if (ch < 3) {
#pragma unroll
        for (int r = 0; r < 8; ++r) {
            const int m = r + (lo ? 0 : 8);
            out[(pixBase + m) * 3 + ch] = acc1[r];
            out[(pixBase + 16 + m) * 3 + ch] = acc2[r];
        }
    }
}

extern "C" void kernel_launch(void* const* d_in, const int* in_sizes, int n_in,
                              void* d_out, int out_size, void* d_ws, size_t ws_size,
                              hipStream_t stream) {
    const float* means3d = (const float*)d_in[0];
    const float* scales  = (const float*)d_in[1];
    const float* opac    = (const float*)d_in[2];
    const float* colors  = (const float*)d_in[3];
    float* ws  = (float*)d_ws;
    float* out = (float*)d_out;

    hipLaunchKernelGGL(dgr_prep_kernel, dim3(1), dim3(512), 0, stream,
                       means3d, scales, opac, colors, ws);

    // 65536 pixels / (256 threads = 8 waves x 32 pixels) = 256 blocks
    hipLaunchKernelGGL(dgr_raster_kernel, dim3(256), dim3(256), 0, stream,
                       ws, out);
}